// GaussianVoxelizer_74887049773608
// MI455X (gfx1250) — compile-verified
//
#include <hip/hip_runtime.h>
#include <hip/hip_bf16.h>
#include <math.h>

typedef float v2f __attribute__((ext_vector_type(2)));
typedef float v8f __attribute__((ext_vector_type(8)));

// D = A(16x4,f32) * B(4x16,f32) + C(16x16,f32), wave32, 8-arg form:
// (neg_a, A, neg_b, B, c_mod, C, reuse_a, reuse_b)
#define WMMA_F32_K4(A, B, C) \
    __builtin_amdgcn_wmma_f32_16x16x4_f32(false, (A), false, (B), (short)0, (C), false, false)

// ---------------------------------------------------------------------------
// Pre-pass: one thread per Gaussian. Invert 3x3 covariance (adjugate),
// apply the 3-sigma volume mask, and emit 12 features already scaled by -0.5
// with ln(masked opacity) folded into the constant term, so the main kernel
// computes density = exp(dot(featA, featB)) with zero epilogue work.
// Feature k pairs with voxel feature:
//   k0..2 : -.5*P00,-.5*P11,-.5*P22      <-> x^2, y^2, z^2
//   k3..5 : -P01, -P02, -P12             <-> xy, xz, yz      (-.5 * 2*Pij)
//   k6..8 : (P*mu)_x,y,z                 <-> x, y, z         (-.5 * -2*(P mu))
//   k9    : -.5*mu'P mu + ln(wop)        <-> 1
//   k10,11: 0 (K padding)                <-> 0
// ---------------------------------------------------------------------------
__global__ void gv_prep(const float* __restrict__ means,
                        const float* __restrict__ opac,
                        const float* __restrict__ cov,
                        const float* __restrict__ vr,
                        float* __restrict__ feat,
                        int G, int Gpad)
{
    int g = blockIdx.x * blockDim.x + threadIdx.x;
    if (g >= Gpad) return;
    float* F = feat + 12 * g;

    if (g >= G) {  // padded rows contribute exp(-inf) = 0
        F[0]=F[1]=F[2]=F[3]=F[4]=F[5]=F[6]=F[7]=F[8]=0.0f;
        F[9] = -INFINITY; F[10]=0.0f; F[11]=0.0f;
        return;
    }

    float mx = means[3*g+0], my = means[3*g+1], mz = means[3*g+2];
    const float* C = cov + 9*g;
    float c00=C[0], c01=C[1], c02=C[2], c11=C[4], c12=C[5], c22=C[8];

    // 3-sigma AABB mask against the volume range
    float s0 = sqrtf(c00), s1 = sqrtf(c11), s2 = sqrtf(c22);
    bool mask = (mx + 3.0f*s0 > vr[0]) && (my + 3.0f*s1 > vr[1]) && (mz + 3.0f*s2 > vr[2]) &&
                (mx - 3.0f*s0 < vr[3]) && (my - 3.0f*s1 < vr[4]) && (mz - 3.0f*s2 < vr[5]);
    float wop = mask ? opac[g] : 0.0f;

    // symmetric 3x3 inverse via adjugate
    float A00 = c11*c22 - c12*c12;
    float A01 = c02*c12 - c01*c22;
    float A02 = c01*c12 - c02*c11;
    float A11 = c00*c22 - c02*c02;
    float A12 = c01*c02 - c00*c12;
    float A22 = c00*c11 - c01*c01;
    float det = c00*A00 + c01*A01 + c02*A02;
    float inv = 1.0f / det;
    float p00=A00*inv, p01=A01*inv, p02=A02*inv;
    float p11=A11*inv, p12=A12*inv, p22=A22*inv;

    float q0 = p00*mx + p01*my + p02*mz;   // (P mu)
    float q1 = p01*mx + p11*my + p12*mz;
    float q2 = p02*mx + p12*my + p22*mz;
    float mpm = mx*q0 + my*q1 + mz*q2;     // mu' P mu

    F[0] = -0.5f * p00;
    F[1] = -0.5f * p11;
    F[2] = -0.5f * p22;
    F[3] = -p01;
    F[4] = -p02;
    F[5] = -p12;
    F[6] = q0;
    F[7] = q1;
    F[8] = q2;
    F[9] = -0.5f * mpm + logf(wop);        // logf(0) = -inf kills masked rows
    F[10] = 0.0f;
    F[11] = 0.0f;
}

// ---------------------------------------------------------------------------
// Main kernel: one wave32 per 16-voxel tile. For each 16-Gaussian tile,
// 3 chained v_wmma_f32_16x16x4_f32 (K=12) compute the 16x16 log-density
// tile; 8 v_exp_f32 + adds accumulate it into the per-voxel sum.
// C/D layout (ISA 7.12.2): lane l holds column N = l%16, rows M = r + 8*(l>=16)
// in VGPR r, so one shfl_xor(16) collapses the Gaussian dimension.
// ---------------------------------------------------------------------------
__global__ __launch_bounds__(256) void gv_voxel(const float* __restrict__ coords,
                                                const float* __restrict__ feat,
                                                float* __restrict__ out,
                                                int ntiles, int gtiles, int nvox)
{
    int wid  = (blockIdx.x * blockDim.x + threadIdx.x) >> 5;   // global wave id = tile id
    if (wid >= ntiles) return;                                 // whole-wave uniform exit
    int lane = threadIdx.x & 31;
    int col  = lane & 15;      // N within tile (and M-row source for A loads)
    int hi   = lane >> 4;      // 0: K=0..1 / rows 0..7   1: K=2..3 / rows 8..15
    int kb   = 2 * hi;

    int v  = wid * 16 + col;
    int vc = v < nvox ? v : nvox - 1;                          // clamp (tail-safe)
    float x = coords[3*vc+0], y = coords[3*vc+1], z = coords[3*vc+2];

    // voxel feature vector, padded to K=12
    float vf[12] = { x*x, y*y, z*z, x*y, x*z, y*z, x, y, z, 1.0f, 0.0f, 0.0f };

    // B fragments per ISA f32 4x16 layout: VGPR0 = K = kb, VGPR1 = K = kb+1
    v2f b0 = { vf[0 + kb], vf[1 + kb] };
    v2f b1 = { vf[4 + kb], vf[5 + kb] };
    v2f b2 = { vf[8 + kb], vf[9 + kb] };

    float acc = 0.0f;
    for (int gt = 0; gt < gtiles; ++gt) {
        const float* F = feat + 12 * (gt * 16 + col) + kb;     // 8B-aligned
        v2f a0 = { F[0], F[1] };
        v2f a1 = { F[4], F[5] };
        v2f a2 = { F[8], F[9] };

        v8f c = {};
        c = WMMA_F32_K4(a0, b0, c);
        c = WMMA_F32_K4(a1, b1, c);
        c = WMMA_F32_K4(a2, b2, c);

        #pragma unroll
        for (int r = 0; r < 8; ++r)
            acc += __expf(c[r]);                               // v_exp_f32 per row
    }

    // collapse Gaussian rows 0..7 (lanes 0-15) + 8..15 (lanes 16-31)
    acc += __shfl_xor(acc, 16, 32);
    if (hi == 0 && v < nvox)
        out[v] = acc;
}

extern "C" void kernel_launch(void* const* d_in, const int* in_sizes, int n_in,
                              void* d_out, int out_size, void* d_ws, size_t ws_size,
                              hipStream_t stream) {
    const float* means  = (const float*)d_in[0];   // (G,3)
    const float* opac   = (const float*)d_in[1];   // (G,)
    const float* cov    = (const float*)d_in[2];   // (G,3,3)
    const float* coords = (const float*)d_in[3];   // (100,100,8,3)
    const float* vr     = (const float*)d_in[4];   // (6,)
    float* out  = (float*)d_out;                   // (100,100,8,1) f32
    float* feat = (float*)d_ws;                    // Gpad*12 floats (48 KB)

    int G    = in_sizes[1];                        // 1024
    int Gpad = (G + 15) & ~15;
    int nvox = out_size;                           // 80000

    gv_prep<<<(Gpad + 255) / 256, 256, 0, stream>>>(means, opac, cov, vr, feat, G, Gpad);

    int ntiles = (nvox + 15) / 16;                 // 5000 (exact)
    int gtiles = Gpad / 16;                        // 64
    int blocks = (ntiles + 7) / 8;                 // 8 waves/block
    gv_voxel<<<blocks, 256, 0, stream>>>(coords, feat, out, ntiles, gtiles, nvox);
}